// GAT_68642167324863
// MI455X (gfx1250) — compile-verified
//
#include <hip/hip_runtime.h>
#include <hip/hip_bf16.h>
#include <cstddef>
#include <cstdint>

// ---------------------------------------------------------------------------
// Fused dense-GAT for MI455X (gfx1250, wave32).
//
// Reference shapes: B=1, N=2048, H=3, E=3.
//   layer0:  c=3, no node term
//   layer1-3: c=3, node term (x: [N,9])
//   layer4:  c=1, node term, head-mean, then softmax over batch axis (B==1)
//
// Strategy (memory-bound workload, 192MB L2 holds edge+adj):
//  * gat_row_kernel: 1 workgroup (256 thr, 8 waves) per destination row i.
//      - stage 24KB edge row + adj row in LDS (async-to-LDS b128 when the
//        toolchain exposes the gfx1250 builtins, float4 copy otherwise)
//      - pass 1: logits (3x3 matvec on raw edge feats via folded G), leaky-relu,
//                masked max  (deterministic LDS tree reduction)
//      - pass 2: w = exp(l-m); accumulate Z[h], S[h][k]=sum w*f, T[h][c]=sum w*n
//                (deterministic LDS tree reduction, no float atomics)
//      - finalize: out = (We*S)/Z + be + T/Z, prelu, write
//  * node_proj_kernel: n = x @ Wn^T + bn via V_WMMA_F32_16X16X4_F32
//      (16 rows / wave, zero-padded LDS operand tiles -> branch-free
//       fragment gathers), exact fp32; then li = a1*n, lj = a2*n.
// ---------------------------------------------------------------------------

typedef __attribute__((ext_vector_type(2))) float v2f;
typedef __attribute__((ext_vector_type(8))) float v8f;

#define NN 2048   // nodes
#define NH 3      // heads

// ---- optional gfx1250 async global->LDS path (guarded) ---------------------
#if defined(__has_builtin)
#  if __has_builtin(__builtin_amdgcn_global_load_async_to_lds_b128) && \
      __has_builtin(__builtin_amdgcn_s_wait_asynccnt)
#    define GAT_HAVE_ASYNC_LDS 1
#  endif
#endif
#ifndef GAT_HAVE_ASYNC_LDS
#  define GAT_HAVE_ASYNC_LDS 0
#endif

#if GAT_HAVE_ASYNC_LDS
// Diagnostic-derived builtin parameter types:
//   arg0: 'int __attribute__((vector_size(16))) __device__ *'  (AS1 = global)
//   arg1: LDS side, same pointee in AS3
typedef int v4i_vs __attribute__((vector_size(16)));
typedef __attribute__((address_space(1))) v4i_vs* gbl_v4i_ptr;
typedef __attribute__((address_space(3))) v4i_vs* lds_v4i_ptr;

__device__ __forceinline__ void gat_async_b128(const void* gsrc, void* lds_dst) {
  // LDS byte offset == low 32 bits of the generic address (aperture rule).
  gbl_v4i_ptr g = (gbl_v4i_ptr)(unsigned long long)(uintptr_t)gsrc;
  lds_v4i_ptr l = (lds_v4i_ptr)(uint32_t)(uintptr_t)lds_dst;
  __builtin_amdgcn_global_load_async_to_lds_b128(g, l, 0, 0);
}
#endif

// ---------------------------------------------------------------------------
// Node projection with fp32 WMMA:  n[j, nn] = sum_k x[j,k] * Wn[nn,k] + bn[nn]
// Also emits li[j,h] = sum_c a1[h,c]*n[j,h*CDIM+c], lj[j,h] likewise with a2.
// NC = 3*CDIM output channels (9 or 3), K fixed at 9 (zero-padded to 12).
// ---------------------------------------------------------------------------
template<int NC, int CDIM>
__global__ __launch_bounds__(32)
void node_proj_kernel(const float* __restrict__ x,     // [NN, 9]
                      const float* __restrict__ Wn,    // [NC, 9]
                      const float* __restrict__ bn,    // [NC]
                      const float* __restrict__ a,     // [3, astride]
                      int astride,
                      float* __restrict__ n_buf,       // [NN, NC]
                      float* __restrict__ li_buf,      // [NN, 3]
                      float* __restrict__ lj_buf)      // [NN, 3]
{
  const int lane    = threadIdx.x;
  const int half    = lane >> 4;     // 0: K pair {k0,k0+1}, 1: {k0+2,k0+3}
  const int m       = lane & 15;     // A-row (M) / B-column (Nout)
  const int rowBase = blockIdx.x * 16;

  __shared__ float xs [16 * 12];     // A tile: 16 rows x K(9->12, zero pad)
  __shared__ float wsb[16 * 12];     // B tile: Nout(9->16) x K(9->12) zero pad
  __shared__ float tile[16][17];     // result staging (+1 pad vs conflicts)

  // zero-fill pads, then write the valid region (branch-free WMMA gathers).
  for (int idx = lane; idx < 16 * 12; idx += 32) { xs[idx] = 0.0f; wsb[idx] = 0.0f; }
  __syncthreads();
  for (int idx = lane; idx < 16 * 9; idx += 32) {
    const int r = idx / 9, k = idx - r * 9;
    xs[r * 12 + k] = x[(rowBase + r) * 9 + k];
  }
  for (int idx = lane; idx < NC * 9; idx += 32) {
    const int nn = idx / 9, k = idx - nn * 9;
    wsb[nn * 12 + k] = Wn[idx];
  }
  __syncthreads();

  // A: 16x4 fp32 (lanes 0-15: K=k0,k0+1 ; lanes 16-31: K=k0+2,k0+3)
  // B: 4x16 fp32 (same striping, columns = output channel)
  v8f acc = {};
#pragma unroll
  for (int s = 0; s < 3; ++s) {
    const int k = 4 * s + 2 * half;
    v2f A = { xs [m * 12 + k], xs [m * 12 + k + 1] };
    v2f B = { wsb[m * 12 + k], wsb[m * 12 + k + 1] };
    // D = A x B + C   (16x16x4, fp32 in/out -> exact math)
    acc = __builtin_amdgcn_wmma_f32_16x16x4_f32(
        /*neg_a=*/false, A, /*neg_b=*/false, B,
        /*c_mod=*/(short)0, acc, /*reuse_a=*/false, /*reuse_b=*/false);
  }

  // C/D layout: VGPR r holds D[M = r + 8*half, N = lane&15]
#pragma unroll
  for (int r = 0; r < 8; ++r) {
    const int M = r + 8 * half;
    float v = acc[r] + ((m < NC) ? bn[m] : 0.0f);
    tile[M][m] = v;
    if (m < NC) n_buf[(size_t)(rowBase + M) * NC + m] = v;
  }
  __syncthreads();

  if (lane < 16) {
    const int row = rowBase + lane;
#pragma unroll
    for (int h = 0; h < NH; ++h) {
      float li = 0.0f, lj = 0.0f;
#pragma unroll
      for (int c = 0; c < CDIM; ++c) {
        const float nv = tile[lane][h * CDIM + c];
        li += a[h * astride + c]        * nv;   // a1
        lj += a[h * astride + CDIM + c] * nv;   // a2
      }
      li_buf[row * 3 + h] = li;
      lj_buf[row * 3 + h] = lj;
    }
  }
}

// ---------------------------------------------------------------------------
// Fused per-row GAT layer. One block = one destination row i.
// CDIM = per-head value width (3 for layers 0-3, 1 for layer 4).
// ---------------------------------------------------------------------------
template<int CDIM, bool HAS_NODE, bool IS_FINAL>
__global__ __launch_bounds__(256)
void gat_row_kernel(const float* __restrict__ edge,    // [NN, NN, 3]
                    const int*   __restrict__ adj,     // [NN, NN]
                    const float* __restrict__ We,      // [3*CDIM, 3]
                    const float* __restrict__ be,      // [3*CDIM]
                    const float* __restrict__ a,       // [3, astride]
                    int astride, int a3off,
                    const float* __restrict__ li_buf,  // [NN,3] or null
                    const float* __restrict__ lj_buf,  // [NN,3] or null
                    const float* __restrict__ n_buf,   // [NN,3*CDIM] or null
                    const float* __restrict__ prelu_w, // [1] or null
                    float* __restrict__ out)           // [NN,3*CDIM] or [NN]
{
  constexpr int NC = 3 * CDIM;
  const int tid = threadIdx.x;
  const int i   = blockIdx.x;

  __shared__ float fsh[NN * 3];          // staged edge row     (24 KB)
  __shared__ float lsh[NN * 3];          // logits / partials   (24 KB)
  __shared__ unsigned char adjsh[NN];    // mask                ( 2 KB)
  __shared__ float red[256 * 3];         // max reduction       ( 3 KB)
  __shared__ float finsh[8];             // final per-head vals

  // Fold a3 into the raw-edge-feature basis:
  //   logit_edge[h] = G[h].f + gb[h], with
  //   G[h][k] = sum_c a3[h,c] * We[h*CDIM+c, k],  gb[h] = sum_c a3[h,c]*be[..]
  float G[3][3], gb[3];
#pragma unroll
  for (int h = 0; h < NH; ++h) {
    G[h][0] = G[h][1] = G[h][2] = 0.0f;
    gb[h] = 0.0f;
#pragma unroll
    for (int c = 0; c < CDIM; ++c) {
      const float a3 = a[h * astride + a3off + c];
#pragma unroll
      for (int k = 0; k < 3; ++k) G[h][k] += a3 * We[(h * CDIM + c) * 3 + k];
      gb[h] += a3 * be[h * CDIM + c];
    }
  }

  // Stage edge row + adj row; warm L2/L0 for the n-buffer gathers.
  {
    const float4* src = reinterpret_cast<const float4*>(edge + (size_t)i * (NN * 3));
    float4* dst = reinterpret_cast<float4*>(fsh);
#if GAT_HAVE_ASYNC_LDS
    for (int idx = tid; idx < (NN * 3) / 4; idx += 256)
      gat_async_b128(src + idx, dst + idx);      // global_load_async_to_lds_b128
#else
    for (int idx = tid; idx < (NN * 3) / 4; idx += 256) dst[idx] = src[idx];
#endif
    const int* arow = adj + (size_t)i * NN;
    for (int j = tid; j < NN; j += 256) adjsh[j] = (unsigned char)(arow[j] != 0);
    if (HAS_NODE) {
      for (int j = tid; j < NN; j += 256)
        __builtin_prefetch(&n_buf[(size_t)j * NC], 0, 1);   // global_prefetch_b8
    }
  }

  float lih[3] = {0.0f, 0.0f, 0.0f};
  if (HAS_NODE) {
#pragma unroll
    for (int h = 0; h < NH; ++h) lih[h] = li_buf[i * 3 + h];
  }
#if GAT_HAVE_ASYNC_LDS
  __builtin_amdgcn_s_wait_asynccnt(0);           // own wave's async copies done
#endif
  __syncthreads();

  // ---- pass 1: logits + masked max -------------------------------------
  float tmax[3] = {-3.0e38f, -3.0e38f, -3.0e38f};
  for (int t = 0; t < NN / 256; ++t) {
    const int j = tid + 256 * t;
    const float f0 = fsh[j * 3 + 0];
    const float f1 = fsh[j * 3 + 1];
    const float f2 = fsh[j * 3 + 2];
    const bool msk = adjsh[j] != 0;
#pragma unroll
    for (int h = 0; h < NH; ++h) {
      float l = G[h][0] * f0 + G[h][1] * f1 + G[h][2] * f2 + gb[h];
      if (HAS_NODE) l += lih[h] + lj_buf[j * 3 + h];
      l = (l >= 0.0f) ? l : 0.2f * l;           // leaky_relu (before mask)
      lsh[j * 3 + h] = l;
      if (msk) tmax[h] = fmaxf(tmax[h], l);
    }
  }
#pragma unroll
  for (int h = 0; h < NH; ++h) red[tid * 3 + h] = tmax[h];
  __syncthreads();
  for (int off = 128; off > 0; off >>= 1) {
    if (tid < off) {
#pragma unroll
      for (int h = 0; h < NH; ++h)
        red[tid * 3 + h] = fmaxf(red[tid * 3 + h], red[(tid + off) * 3 + h]);
    }
    __syncthreads();
  }
  const float mh[3] = {red[0], red[1], red[2]};

  // ---- pass 2: weighted accumulation -----------------------------------
  float Z[3] = {0.0f, 0.0f, 0.0f};
  float S[3][3] = {};
  float T[3][CDIM] = {};
  for (int t = 0; t < NN / 256; ++t) {
    const int j = tid + 256 * t;
    if (!adjsh[j]) continue;                    // masked -> weight ~ exp(-9e15)
    const float f0 = fsh[j * 3 + 0];
    const float f1 = fsh[j * 3 + 1];
    const float f2 = fsh[j * 3 + 2];
#pragma unroll
    for (int h = 0; h < NH; ++h) {
      const float w = __expf(lsh[j * 3 + h] - mh[h]);
      Z[h]    += w;
      S[h][0] += w * f0;
      S[h][1] += w * f1;
      S[h][2] += w * f2;
      if (HAS_NODE) {
#pragma unroll
        for (int c = 0; c < CDIM; ++c)
          T[h][c] += w * n_buf[(size_t)j * NC + h * CDIM + c];
      }
    }
  }
  __syncthreads();                 // lsh free now; reuse for partials

  // Deterministic tree reduction of (Z, S, T): 24 slots/thread, stride 24.
  float* part = lsh;               // 256*24 = 6144 floats == sizeof(lsh)
#pragma unroll
  for (int h = 0; h < NH; ++h) {
    part[tid * 24 + h] = Z[h];
#pragma unroll
    for (int k = 0; k < 3; ++k) part[tid * 24 + 3 + h * 3 + k] = S[h][k];
#pragma unroll
    for (int c = 0; c < CDIM; ++c) part[tid * 24 + 12 + h * CDIM + c] = T[h][c];
  }
  __syncthreads();
  for (int off = 128; off > 0; off >>= 1) {
    if (tid < off) {
#pragma unroll
      for (int v = 0; v < 12 + 3 * CDIM; ++v)
        part[tid * 24 + v] += part[(tid + off) * 24 + v];
    }
    __syncthreads();
  }

  // ---- finalize ---------------------------------------------------------
  // out[i,h,c] = (We[nn,:].S[h,:])/Z[h] + be[nn] + T[h,c]/Z[h]
  if (tid < NC) {
    const int h = tid / CDIM;
    const float z = part[h];
    float val = 0.0f;
#pragma unroll
    for (int k = 0; k < 3; ++k) val += We[tid * 3 + k] * part[3 + h * 3 + k];
    val /= z;
    val += be[tid];
    if (HAS_NODE) val += part[12 + tid] / z;
    if (!IS_FINAL) {
      const float pw = prelu_w[0];
      val = (val >= 0.0f) ? val : pw * val;     // prelu
      out[(size_t)i * NC + tid] = val;
    } else {
      finsh[tid] = val;                          // CDIM==1 -> tid == h
    }
  }
  if (IS_FINAL) {
    __syncthreads();
    if (tid == 0) {
      const float mean = (finsh[0] + finsh[1] + finsh[2]) * (1.0f / 3.0f);
      // softmax over batch axis with B==1: exp(v - v) / exp(v - v) == 1
      out[i] = __expf(mean - mean);
    }
  }
}

// ---------------------------------------------------------------------------
// Host launch. Input flattening order (setup_inputs dict order):
//  [0] edge_feats  [1] adj
//  l0: [2]We [3]be [4]a
//  l1: [5]Wn [6]bn [7]We [8]be [9]a
//  l2: [10..14]   l3: [15..19]
//  l4: [20]Wn [21]bn [22]We [23]be [24]a
//  [25..28] p0..p3
// ---------------------------------------------------------------------------
extern "C" void kernel_launch(void* const* d_in, const int* in_sizes, int n_in,
                              void* d_out, int out_size, void* d_ws, size_t ws_size,
                              hipStream_t stream) {
  (void)in_sizes; (void)n_in; (void)out_size; (void)ws_size;

  const float* edge = (const float*)d_in[0];
  const int*   adj  = (const int*)d_in[1];
  auto F = [&](int idx) { return (const float*)d_in[idx]; };

  float* ws = (float*)d_ws;
  float* xA = ws;                       // [NN*9]
  float* xB = ws + NN * 9;              // [NN*9]
  float* nb = ws + 2 * NN * 9;          // [NN*9]
  float* li = ws + 3 * NN * 9;          // [NN*3]
  float* lj = li + NN * 3;              // [NN*3]

  const dim3 gRow(NN), bRow(256);
  const dim3 gProj(NN / 16), bProj(32);

  // ---- layer 0 (no node term), prelu p0 -> xA
  gat_row_kernel<3, false, false><<<gRow, bRow, 0, stream>>>(
      edge, adj, F(2), F(3), F(4), /*astride=*/3, /*a3off=*/0,
      nullptr, nullptr, nullptr, F(25), xA);

  // ---- layers 1..3, prelu p1..p3
  float* xin = xA;
  float* xout = xB;
  for (int L = 0; L < 3; ++L) {
    const int b = 5 + L * 5;
    node_proj_kernel<9, 3><<<gProj, bProj, 0, stream>>>(
        xin, F(b + 0), F(b + 1), F(b + 4), /*astride=*/9, nb, li, lj);
    gat_row_kernel<3, true, false><<<gRow, bRow, 0, stream>>>(
        edge, adj, F(b + 2), F(b + 3), F(b + 4), /*astride=*/9, /*a3off=*/6,
        li, lj, nb, F(26 + L), xout);
    float* t = xin; xin = xout; xout = t;
  }

  // ---- layer 4 (c=1, head-mean, batch softmax over B==1)
  node_proj_kernel<3, 1><<<gProj, bProj, 0, stream>>>(
      xin, F(20), F(21), F(24), /*astride=*/3, nb, li, lj);
  gat_row_kernel<1, true, true><<<gRow, bRow, 0, stream>>>(
      edge, adj, F(22), F(23), F(24), /*astride=*/3, /*a3off=*/2,
      li, lj, nb, nullptr, (float*)d_out);
}